// BiasedMpcController_29678224015736
// MI455X (gfx1250) — compile-verified
//
#include <hip/hip_runtime.h>
#include <math.h>

// Problem dimensions (fixed by the reference)
#define BATCH   128
#define Hh      12
#define XD      8
#define UD      4
#define SD      4
#define NXx     96      // H * x_dim
#define NN      240     // H * (2*x_dim + u_dim)
#define MM      192     // 2 * NX (inequality rows)
#define NEQ     96      // equality rows
#define ITERS   20
#define NTHREADS 256
#define NWAVES  8
#define SSTRIDE 97      // 96 + 1 pad (S matrix)
#define AESTR   241     // 240 + 1 pad (Aeq rows) -> column reads hit distinct banks

typedef __attribute__((ext_vector_type(2))) float v2f;
typedef __attribute__((ext_vector_type(8))) float v8f;

// Aeq[i,j] from structure: Aeq = [L, L, -B_aug], L = I - kron(subdiag, A).
// Used ONCE at kernel start to densify Aeq into LDS; the WMMA loop is branch-free.
__device__ __forceinline__ float aeq_el(int i, int j,
                                        const float* __restrict__ sAm,
                                        const float* __restrict__ sBm) {
  int ti = i >> 3, r = i & 7;
  if (j < MM) {
    int jj = (j >= NXx) ? (j - NXx) : j;
    int tj = jj >> 3, c = jj & 7;
    if (tj == ti)     return (c == r) ? 1.0f : 0.0f;
    if (tj == ti - 1) return -sAm[r * XD + c];
    return 0.0f;
  }
  int ju = j - MM;
  int tu = ju >> 2, cu = ju & 3;
  return (tu == ti) ? -sBm[r * UD + cu] : 0.0f;
}

__global__ __launch_bounds__(NTHREADS)
void BiasedMpcController_ipm_kernel(const float* __restrict__ x0,
                                    const float* __restrict__ forecast,
                                    const float* __restrict__ Amat,
                                    const float* __restrict__ Bmat,
                                    const float* __restrict__ Cmat,
                                    const float* __restrict__ Qh,
                                    const float* __restrict__ Qb,
                                    const float* __restrict__ Rm,
                                    float* __restrict__ out)
{
  __shared__ float sAeq[NEQ * AESTR];          // dense Aeq, 96 x 240 (padded), 92.5 KB
  __shared__ float sS[NXx * SSTRIDE];          // Schur complement, 96x96 (padded)
  __shared__ float sAm[64], sBm[32], sCm[32];
  __shared__ float qd[NN], dinv[NN], bv[NEQ];
  __shared__ float z[NN], sv[MM], lm[MM], nu[NEQ];
  __shared__ float rp1[MM], rhs[NN];
  __shared__ float dzv[NN], dsv[MM], dlv[MM], dnuv[NEQ];
  __shared__ float red[NTHREADS];
  __shared__ float mu_s, alpha_s;

  const int tid  = threadIdx.x;
  const int bid  = blockIdx.x;                  // one workgroup per batch element
  const int lane = tid & 31;
  const int wid  = tid >> 5;

  // ---- stage tiny constant matrices + diag(Qm), init IPM state ----
  if (tid < 64)                 sAm[tid]      = Amat[tid];
  if (tid >= 64 && tid < 96)    sBm[tid - 64] = Bmat[tid - 64];
  if (tid >= 96 && tid < 128)   sCm[tid - 96] = Cmat[tid - 96];
  for (int j = tid; j < NN; j += NTHREADS) {
    float qv;
    if (j < NXx)        qv = Qh[9 * (j & 7)];   // diag of 8x8
    else if (j < MM)    qv = Qb[9 * (j & 7)];
    else                qv = Rm[5 * (j & 3)];   // diag of 4x4
    qd[j] = qv;
    z[j]  = 0.0f;
  }
  for (int i = tid; i < MM;  i += NTHREADS) { sv[i] = 1.0f; lm[i] = 1.0f; }
  for (int i = tid; i < NEQ; i += NTHREADS) nu[i] = 0.0f;
  __syncthreads();

  // ---- densify Aeq into LDS once (iteration-invariant) ----
  for (int idx = tid; idx < NEQ * NN; idx += NTHREADS) {
    int i = idx / NN, k = idx - i * NN;
    sAeq[i * AESTR + k] = aeq_el(i, k, sAm, sBm);
  }

  // b = C_aug @ forecast_flat + l_aug @ x0
  for (int i = tid; i < NEQ; i += NTHREADS) {
    int ti = i >> 3, r = i & 7;
    float acc = 0.0f;
    for (int s2 = 0; s2 < SD; ++s2)
      acc += sCm[r * SD + s2] * forecast[(bid * Hh + ti) * SD + s2];
    if (ti == 0)
      for (int c = 0; c < XD; ++c)
        acc += sAm[r * XD + c] * x0[bid * XD + c];
    bv[i] = acc;
  }
  __syncthreads();

  // ================= IPM iterations =================
  for (int it = 0; it < ITERS; ++it) {
    // --- mu = mean(s .* lam) ---
    float loc = 0.0f;
    for (int i = tid; i < MM; i += NTHREADS) loc += sv[i] * lm[i];
    red[tid] = loc; __syncthreads();
    for (int off = NTHREADS / 2; off > 0; off >>= 1) {
      if (tid < off) red[tid] += red[tid + off];
      __syncthreads();
    }
    if (tid == 0) mu_s = red[0] / (float)MM;

    // --- D^{-1} = 1/(q + lam/s on first 2*NX entries) ---
    for (int j = tid; j < NN; j += NTHREADS)
      dinv[j] = 1.0f / (qd[j] + ((j < MM) ? lm[j] / sv[j] : 0.0f));

    // --- r_p1 = Gz + s ---
    for (int i = tid; i < MM; i += NTHREADS)
      rp1[i] = ((i < NXx) ? -z[i] : z[i]) + sv[i];
    __syncthreads();
    const float mu = mu_s;

    // --- rhs_z = -(r_d + ((-r_c + lam*r_p1)/s) @ G) ; Aeq^T nu inlined ---
    for (int j = tid; j < NN; j += NTHREADS) {
      float atn;
      if (j < MM) {
        int jj = (j >= NXx) ? j - NXx : j;
        int tj = jj >> 3, c = jj & 7;
        atn = nu[jj];
        if (tj < Hh - 1) {
          float a2 = 0.0f;
          for (int r = 0; r < XD; ++r) a2 += sAm[r * XD + c] * nu[(tj + 1) * XD + r];
          atn -= a2;
        }
      } else {
        int ju = j - MM, tu = ju >> 2, cu = ju & 3;
        float a2 = 0.0f;
        for (int r = 0; r < XD; ++r) a2 += sBm[r * UD + cu] * nu[tu * XD + r];
        atn = -a2;
      }
      float gl = 0.0f, tg = 0.0f;
      if (j < MM) {
        float rc = lm[j] * sv[j] - 0.1f * mu;
        float tt = (-rc + lm[j] * rp1[j]) / sv[j];
        gl = (j < NXx) ? -lm[j] : lm[j];
        tg = (j < NXx) ? -tt : tt;
      }
      rhs[j] = -(qd[j] * z[j] + gl + atn + tg);
    }
    __syncthreads();

    // --- Schur RHS: g = Aeq (D^{-1} rhs) + (Aeq z - b)  -> dnuv ---
    for (int i = tid; i < NEQ; i += NTHREADS) {
      int ti = i >> 3, r = i & 7;
      float acc = dinv[i] * rhs[i] + dinv[NXx + i] * rhs[NXx + i]
                + z[i] + z[NXx + i] - bv[i];
      if (ti > 0) {
        for (int c = 0; c < XD; ++c) {
          int jh = (ti - 1) * XD + c;
          acc -= sAm[r * XD + c] *
                 (dinv[jh] * rhs[jh] + dinv[NXx + jh] * rhs[NXx + jh]
                  + z[jh] + z[NXx + jh]);
        }
      }
      for (int cu = 0; cu < UD; ++cu) {
        int ju = MM + ti * UD + cu;
        acc -= sBm[r * UD + cu] * (dinv[ju] * rhs[ju] + z[ju]);
      }
      dnuv[i] = acc;
    }

    // --- S = (Aeq D^{-1}) @ Aeq^T via V_WMMA_F32_16X16X4_F32 ---
    // Branch-free K-loop: fragments come straight from LDS (dense Aeq),
    // with D^{-1} folded into the A operand by one v_mul each.
    {
      const int mrow = lane & 15;
      const int half = lane >> 4;            // 0 or 1
      for (int tile = wid; tile < 36; tile += NWAVES) {
        const int r0 = (tile / 6) * 16, c0 = (tile % 6) * 16;
        const float* arow = &sAeq[(r0 + mrow) * AESTR];   // A-fragment source row
        const float* brow = &sAeq[(c0 + mrow) * AESTR];   // B-fragment source row (Aeq^T)
        v8f acc = {};
        #pragma unroll 4
        for (int k0 = 0; k0 < NN; k0 += 4) {
          // A (16x4 f32): VGPR0 -> K = k0+2*half, VGPR1 -> K = k0+2*half+1
          int ka = k0 + 2 * half;
          v2f af, bf;
          af.x = arow[ka]     * dinv[ka];
          af.y = arow[ka + 1] * dinv[ka + 1];
          // B (4x16 f32): VGPR0 -> K = k0+half, VGPR1 -> K = k0+half+2
          int kb = k0 + half;
          bf.x = brow[kb];
          bf.y = brow[kb + 2];
          acc = __builtin_amdgcn_wmma_f32_16x16x4_f32(
                  /*neg_a=*/false, af, /*neg_b=*/false, bf,
                  /*c_mod=*/(short)0, acc, /*reuse_a=*/false, /*reuse_b=*/false);
        }
        #pragma unroll
        for (int v = 0; v < 8; ++v) {
          int mm2 = v + half * 8;            // C/D layout: M = v (+8 for lanes 16-31)
          sS[(r0 + mm2) * SSTRIDE + (c0 + mrow)] = acc[v];
        }
      }
    }
    __syncthreads();

    // --- Cholesky of S (lower, in place) ---
    for (int k = 0; k < NEQ; ++k) {
      if (tid == 0) sS[k * SSTRIDE + k] = sqrtf(fmaxf(sS[k * SSTRIDE + k], 1e-20f));
      __syncthreads();
      float dkk = sS[k * SSTRIDE + k];
      for (int i = k + 1 + tid; i < NEQ; i += NTHREADS) sS[i * SSTRIDE + k] /= dkk;
      __syncthreads();
      for (int idx = tid; idx < NEQ * NEQ; idx += NTHREADS) {
        int i = idx / NEQ, j = idx - i * NEQ;
        if (j > k && j <= i)
          sS[i * SSTRIDE + j] -= sS[i * SSTRIDE + k] * sS[j * SSTRIDE + k];
      }
      __syncthreads();
    }
    // --- forward solve L y = g (in dnuv) ---
    for (int k = 0; k < NEQ; ++k) {
      if (tid == 0) dnuv[k] /= sS[k * SSTRIDE + k];
      __syncthreads();
      float yk = dnuv[k];
      for (int i = k + 1 + tid; i < NEQ; i += NTHREADS) dnuv[i] -= sS[i * SSTRIDE + k] * yk;
      __syncthreads();
    }
    // --- backward solve L^T dnu = y ---
    for (int k = NEQ - 1; k >= 0; --k) {
      if (tid == 0) dnuv[k] /= sS[k * SSTRIDE + k];
      __syncthreads();
      float xk = dnuv[k];
      for (int i = tid; i < k; i += NTHREADS) dnuv[i] -= sS[k * SSTRIDE + i] * xk;
      __syncthreads();
    }

    // --- dz = D^{-1} (rhs - Aeq^T dnu) ---
    for (int j = tid; j < NN; j += NTHREADS) {
      float atd;
      if (j < MM) {
        int jj = (j >= NXx) ? j - NXx : j;
        int tj = jj >> 3, c = jj & 7;
        atd = dnuv[jj];
        if (tj < Hh - 1) {
          float a2 = 0.0f;
          for (int r = 0; r < XD; ++r) a2 += sAm[r * XD + c] * dnuv[(tj + 1) * XD + r];
          atd -= a2;
        }
      } else {
        int ju = j - MM, tu = ju >> 2, cu = ju & 3;
        float a2 = 0.0f;
        for (int r = 0; r < XD; ++r) a2 += sBm[r * UD + cu] * dnuv[tu * XD + r];
        atd = -a2;
      }
      dzv[j] = dinv[j] * (rhs[j] - atd);
    }
    __syncthreads();

    // --- ds = -r_p1 - dz G^T ;  dlam = (-r_c - lam ds)/s ---
    for (int i = tid; i < MM; i += NTHREADS) {
      float gdz = (i < NXx) ? -dzv[i] : dzv[i];
      float dsi = -rp1[i] - gdz;
      dsv[i] = dsi;
      float rc = lm[i] * sv[i] - 0.1f * mu;
      dlv[i] = (-rc - lm[i] * dsi) / sv[i];
    }
    __syncthreads();

    // --- alpha = min(1, 0.99 * min step-to-boundary) ---
    float amin = 1e10f;
    for (int i = tid; i < MM; i += NTHREADS) {
      if (dsv[i] < 0.0f) amin = fminf(amin, -sv[i] / dsv[i]);
      if (dlv[i] < 0.0f) amin = fminf(amin, -lm[i] / dlv[i]);
    }
    red[tid] = amin; __syncthreads();
    for (int off = NTHREADS / 2; off > 0; off >>= 1) {
      if (tid < off) red[tid] = fminf(red[tid], red[tid + off]);
      __syncthreads();
    }
    if (tid == 0) alpha_s = fminf(1.0f, 0.99f * red[0]);
    __syncthreads();
    const float a = alpha_s;

    for (int j = tid; j < NN;  j += NTHREADS) z[j]  += a * dzv[j];
    for (int i = tid; i < MM;  i += NTHREADS) { sv[i] += a * dsv[i]; lm[i] += a * dlv[i]; }
    for (int i = tid; i < NEQ; i += NTHREADS) nu[i] += a * dnuv[i];
    __syncthreads();
  }

  // ---- outputs: x_sol = z[:96]+z[96:192], u_sol = z[192:240] ----
  for (int i = tid; i < NXx; i += NTHREADS)
    out[bid * NXx + i] = z[i] + z[NXx + i];
  float* uo = out + BATCH * NXx;
  for (int i = tid; i < Hh * UD; i += NTHREADS)
    uo[bid * (Hh * UD) + i] = z[MM + i];
}

extern "C" void kernel_launch(void* const* d_in, const int* in_sizes, int n_in,
                              void* d_out, int out_size, void* d_ws, size_t ws_size,
                              hipStream_t stream) {
  const float* x0       = (const float*)d_in[0];
  const float* forecast = (const float*)d_in[1];
  const float* A        = (const float*)d_in[2];
  const float* B        = (const float*)d_in[3];
  const float* C        = (const float*)d_in[4];
  const float* Qh       = (const float*)d_in[5];
  const float* Qb       = (const float*)d_in[6];
  const float* R        = (const float*)d_in[7];
  float* out = (float*)d_out;
  hipLaunchKernelGGL(BiasedMpcController_ipm_kernel,
                     dim3(BATCH), dim3(NTHREADS), 0, stream,
                     x0, forecast, A, B, C, Qh, Qb, R, out);
  (void)in_sizes; (void)n_in; (void)out_size; (void)d_ws; (void)ws_size;
}